// GATClassifier_54039278518429
// MI455X (gfx1250) — compile-verified
//
#include <hip/hip_runtime.h>
#include <math.h>

#define NN 50000
#define NE 800000
#define INDIM 256
#define HID 32
#define HEADS 4
#define ETOT (NE + NN)

typedef __attribute__((ext_vector_type(2))) float v2f;
typedef __attribute__((ext_vector_type(8))) float v8f;

__device__ __forceinline__ unsigned fkey(float x) {
  unsigned u = __float_as_uint(x);
  return (u & 0x80000000u) ? ~u : (u | 0x80000000u);
}
__device__ __forceinline__ float unfkey(unsigned k) {
  return __uint_as_float((k & 0x80000000u) ? (k & 0x7fffffffu) : ~k);
}
__device__ __forceinline__ float elu(float v) {
  return (v > 0.f) ? v : (__expf(v) - 1.f);
}

__global__ void fill_u32(unsigned* p, unsigned v, int n) {
  int i = blockIdx.x * blockDim.x + threadIdx.x;
  if (i < n) p[i] = v;
}
__global__ void fill_f32(float* p, float v, int n) {
  int i = blockIdx.x * blockDim.x + threadIdx.x;
  if (i < n) p[i] = v;
}

// C[M,N] = A[M,K] @ B[K,N], row-major, exact fp32 via V_WMMA_F32_16X16X4_F32.
// N,K compile-time so all tile offsets fold into 24-bit immediate IOFFSETs.
// block = (32, NT): one wave per 16x16 C tile; M,N multiples of 16, K multiple of 4.
template <int N, int K>
__global__ void gemm_wmma_f32(const float* __restrict__ A, const float* __restrict__ B,
                              float* __restrict__ C) {
  const int lane  = threadIdx.x;                    // 0..31
  const int mbase = blockIdx.x * 16;
  const int nbase = (blockIdx.y * blockDim.y + threadIdx.y) * 16;
  const int l15   = lane & 15;
  const int koff  = (lane >> 4) << 1;               // 0 or 2
  const float* ap = A + (size_t)(mbase + l15) * K + koff;   // row l15, cols koff..koff+1
  const float* bp = B + (size_t)koff * N + nbase + l15;     // rows koff..koff+1, col l15
  v8f c = {0.f, 0.f, 0.f, 0.f, 0.f, 0.f, 0.f, 0.f};
#pragma unroll 8
  for (int k0 = 0; k0 < K; k0 += 4) {
    v2f a, b;
    a.x = ap[k0];
    a.y = ap[k0 + 1];
    b.x = bp[k0 * N];                               // immediate offsets after unroll
    b.y = bp[(k0 + 1) * N];
    c = __builtin_amdgcn_wmma_f32_16x16x4_f32(false, a, false, b, (short)0, c,
                                              false, false);
  }
  const int crow = mbase + ((lane >> 4) << 3);      // +0 or +8
  float* cp = C + (size_t)crow * N + nbase + l15;
#pragma unroll
  for (int r = 0; r < 8; ++r) cp[r * N] = c[r];
}

// as[n,h] = <h[n,h,:], att_s[h,:]>, ad likewise. one thread per (n,h).
__global__ void att_scores(const float* __restrict__ h, const float* __restrict__ att_s,
                           const float* __restrict__ att_d, float* __restrict__ as,
                           float* __restrict__ ad, int H, int C) {
  int i = blockIdx.x * blockDim.x + threadIdx.x;
  if (i >= NN * H) return;
  int hh = i % H;
  const float* hp = h + (size_t)i * C;
  float s = 0.f, d = 0.f;
  for (int c = 0; c < C; ++c) {
    float v = hp[c];
    s += v * att_s[hh * C + c];
    d += v * att_d[hh * C + c];
  }
  as[i] = s;
  ad[i] = d;
}

// segment max of leakyrelu(as[src]+ad[dst]) keyed per (dst,head), via uint atomicMax
__global__ void edge_max(const int* __restrict__ src, const int* __restrict__ dst,
                         const float* __restrict__ as, const float* __restrict__ ad,
                         unsigned* __restrict__ mkey, int H) {
  long long i = (long long)blockIdx.x * blockDim.x + threadIdx.x;
  if (i >= (long long)ETOT * H) return;
  int e = (int)(i / H), h = (int)(i % H);
  int s = (e < NE) ? src[e] : (e - NE);
  int d = (e < NE) ? dst[e] : (e - NE);
  float x = as[s * H + h] + ad[d * H + h];
  x = (x > 0.f) ? x : 0.2f * x;
  atomicMax(&mkey[d * H + h], fkey(x));
}

// acc[dst,h,c] += p * feat[src,h,c];  denom[dst,h] += p.  one thread per (e,h,c).
__global__ void edge_acc(const int* __restrict__ src, const int* __restrict__ dst,
                         const float* __restrict__ as, const float* __restrict__ ad,
                         const unsigned* __restrict__ mkey, const float* __restrict__ feat,
                         float* __restrict__ acc, float* __restrict__ denom, int H, int C) {
  long long i = (long long)blockIdx.x * blockDim.x + threadIdx.x;
  if (i >= (long long)ETOT * H * C) return;
  int c = (int)(i % C);
  long long eh = i / C;
  int h = (int)(eh % H);
  int e = (int)(eh / H);
  int s = (e < NE) ? src[e] : (e - NE);
  int d = (e < NE) ? dst[e] : (e - NE);
  float x = as[s * H + h] + ad[d * H + h];
  x = (x > 0.f) ? x : 0.2f * x;
  float p = __expf(x - unfkey(mkey[d * H + h]));
  atomicAdd(&acc[((size_t)d * H + h) * C + c], p * feat[((size_t)s * H + h) * C + c]);
  if (c == 0) atomicAdd(&denom[d * H + h], p);
}

// acc[n,h,c] = elu(acc/denom + bias) in place (concat layout)
__global__ void norm_elu(float* __restrict__ acc, const float* __restrict__ denom,
                         const float* __restrict__ bias, int H, int C) {
  int i = blockIdx.x * blockDim.x + threadIdx.x;
  if (i >= NN * H * C) return;
  int n = i / (H * C);
  int r = i % (H * C);
  int h = r / C;
  acc[i] = elu(acc[i] / denom[n * H + h] + bias[r]);
}

// per node: v[c]=elu(acc2/denom2+b2); out[n,:] = v @ lin_w + lin_b. 1 wave32 / node.
__global__ void final_cls(const float* __restrict__ acc2, const float* __restrict__ denom2,
                          const float* __restrict__ b2, const float* __restrict__ lin_w,
                          const float* __restrict__ lin_b, float* __restrict__ out) {
  int n = blockIdx.x * (blockDim.x >> 5) + (threadIdx.x >> 5);
  int c = threadIdx.x & 31;
  if (n >= NN) return;
  float v = elu(acc2[(size_t)n * HID + c] / denom2[n] + b2[c]);
  float w0 = v * lin_w[c * 2 + 0];
  float w1 = v * lin_w[c * 2 + 1];
#pragma unroll
  for (int m = 16; m > 0; m >>= 1) {
    w0 += __shfl_xor(w0, m, 32);
    w1 += __shfl_xor(w1, m, 32);
  }
  if (c == 0) {
    out[n * 2 + 0] = w0 + lin_b[0];
    out[n * 2 + 1] = w1 + lin_b[1];
  }
}

static inline unsigned blks(long long tot, int bs) { return (unsigned)((tot + bs - 1) / bs); }

extern "C" void kernel_launch(void* const* d_in, const int* in_sizes, int n_in,
                              void* d_out, int out_size, void* d_ws, size_t ws_size,
                              hipStream_t stream) {
  const float* x        = (const float*)d_in[0];
  const int*   ei       = (const int*)d_in[1];
  const float* W1       = (const float*)d_in[2];
  const float* att_src1 = (const float*)d_in[3];
  const float* att_dst1 = (const float*)d_in[4];
  const float* b1       = (const float*)d_in[5];
  const float* W2       = (const float*)d_in[6];
  const float* att_src2 = (const float*)d_in[7];
  const float* att_dst2 = (const float*)d_in[8];
  const float* b2       = (const float*)d_in[9];
  const float* lin_w    = (const float*)d_in[10];
  const float* lin_b    = (const float*)d_in[11];
  float* out = (float*)d_out;
  const int* src = ei;
  const int* dst = ei + NE;

  // workspace layout (floats)
  float*    ws   = (float*)d_ws;
  float*    h1   = ws;                              // 50000*128 = 6.4M
  float*    as1  = ws + 6400000;                    // 200k
  float*    ad1  = as1 + 200000;                    // 200k
  unsigned* m1   = (unsigned*)(ad1 + 200000);       // 200k
  float*    d1   = (float*)(m1 + 200000);           // 200k
  float*    acc1 = d1 + 200000;                     // 6.4M (becomes h2 after norm)
  // stage 2 aliases inside dead h1 region
  float*    z    = h1;                              // 50000*32 = 1.6M
  float*    acc2 = h1 + 1600000;                    // 1.6M
  float*    as2  = h1 + 3200000;                    // 50k
  float*    ad2  = as2 + 50000;                     // 50k
  unsigned* m2   = (unsigned*)(ad2 + 50000);        // 50k
  float*    d2   = (float*)(m2 + 50000);            // 50k

  // ---- layer 1 ----
  // h1 = x @ W1  [50000x256]@[256x128]; 4 N-tiles per block, 2 y-blocks
  gemm_wmma_f32<HEADS * HID, INDIM>
      <<<dim3(NN / 16, 2), dim3(32, 4), 0, stream>>>(x, W1, h1);
  att_scores<<<blks(NN * HEADS, 256), 256, 0, stream>>>(h1, att_src1, att_dst1, as1, ad1, HEADS, HID);
  fill_u32<<<blks(200000, 256), 256, 0, stream>>>(m1, 0u, 200000);
  fill_f32<<<blks(200000, 256), 256, 0, stream>>>(d1, 0.f, 200000);
  fill_f32<<<blks(6400000, 256), 256, 0, stream>>>(acc1, 0.f, 6400000);
  edge_max<<<blks((long long)ETOT * HEADS, 256), 256, 0, stream>>>(src, dst, as1, ad1, m1, HEADS);
  edge_acc<<<blks((long long)ETOT * HEADS * HID, 256), 256, 0, stream>>>(src, dst, as1, ad1, m1,
                                                                         h1, acc1, d1, HEADS, HID);
  norm_elu<<<blks(6400000, 256), 256, 0, stream>>>(acc1, d1, b1, HEADS, HID);

  // ---- layer 2 ----
  // z = acc1 @ W2  [50000x128]@[128x32]; 2 N-tiles per block
  gemm_wmma_f32<HID, HEADS * HID>
      <<<dim3(NN / 16, 1), dim3(32, 2), 0, stream>>>(acc1, W2, z);
  att_scores<<<blks(NN, 256), 256, 0, stream>>>(z, att_src2, att_dst2, as2, ad2, 1, HID);
  fill_u32<<<blks(50000, 256), 256, 0, stream>>>(m2, 0u, 50000);
  fill_f32<<<blks(50000, 256), 256, 0, stream>>>(d2, 0.f, 50000);
  fill_f32<<<blks(1600000, 256), 256, 0, stream>>>(acc2, 0.f, 1600000);
  edge_max<<<blks((long long)ETOT, 256), 256, 0, stream>>>(src, dst, as2, ad2, m2, 1);
  edge_acc<<<blks((long long)ETOT * HID, 256), 256, 0, stream>>>(src, dst, as2, ad2, m2,
                                                                 z, acc2, d2, 1, HID);

  // ---- classifier (fused with layer-2 normalize/ELU/mean) ----
  final_cls<<<blks((long long)NN * 32, 256), 256, 0, stream>>>(acc2, d2, b2, lin_w, lin_b, out);
}